// STGCN_21620865368175
// MI455X (gfx1250) — compile-verified
//
#include <hip/hip_runtime.h>
#include <hip/hip_bf16.h>

typedef __attribute__((ext_vector_type(2))) float v2f;
typedef __attribute__((ext_vector_type(8))) float v8f;

// D = A(16x4) * B(4x16) + C  on wave32, full-fp32 WMMA.
__device__ __forceinline__ v8f wmma4(v2f a, v2f b, v8f c) {
  return __builtin_amdgcn_wmma_f32_16x16x4_f32(
      /*neg_a=*/false, a, /*neg_b=*/false, b, /*c_mod=*/(short)0, c,
      /*reuse_a=*/false, /*reuse_b=*/false);
}

// Paired-interleaved LDS layout: element (k, j) of a K x J operand lives at
//   (k>>1)*STRIDE + 2*j + (k&1),  STRIDE = 2*J + 2.
// -> each lane's (k, k+1) fragment pair is one aligned b64; k-offsets are
//    affine so unrolled loops use immediate ds offsets; stores conflict-free.
#define STR64 130  // J = 64
#define STR16 34   // J = 16

// ---------------- workspace layout (in floats) ----------------
static const size_t OFF_ADJN  = 0;          // 16*64*64        = 65536
static const size_t OFF_STAT1 = 65536;      // 64 slots * 128  = 8192
static const size_t OFF_STAT2 = 73728;      // 8192
static const size_t OFF_BN1   = 81920;      // 128 (scale, shift)
static const size_t OFF_BN2   = 82048;      // 128
static const size_t OFF_S     = 82176;      // 16*4096 = 65536
static const size_t OFF_F     = 147712;     // 65536
static const size_t OFF_L     = 213248;     // 65536
static const size_t OFF_Z1    = 278784;     // 16*512*4096 = 33554432
static const size_t OFF_Z2    = 33833216;   // 33554432

// ---------------- K0: zero accumulators ----------------
__global__ __launch_bounds__(256) void k_zero(float* __restrict__ p, int n) {
  int i = blockIdx.x * 256 + threadIdx.x;
  if (i < n) p[i] = 0.f;
}

// ---------------- K1: adj_norm = D^-1/2 A D^-1/2 ----------------
__global__ __launch_bounds__(64) void k_adj(const float* __restrict__ adj,
                                            float* __restrict__ adjn) {
  __shared__ float dis[64];
  const int b = blockIdx.x, v = threadIdx.x;
  const float* Ab = adj + b * 4096;
  float s = 0.f;
  for (int w = 0; w < 64; ++w) s += Ab[v * 64 + w];
  dis[v] = rsqrtf(s + 1e-6f);
  __syncthreads();
  const float dv = dis[v];
  float* An = adjn + b * 4096;
  for (int w = 0; w < 64; ++w) An[v * 64 + w] = dv * Ab[v * 64 + w] * dis[w];
}

// C-tile emit: write z, accumulate per-channel sum/sumsq partials.
__device__ __forceinline__ void emit_tile(v8f c, float* __restrict__ zs,
                                          float* __restrict__ st, int rbase, int ccol) {
  float s = 0.f, q = 0.f;
#pragma unroll
  for (int r = 0; r < 8; ++r) {
    float v = c[r];
    zs[((rbase + r) << 6) + ccol] = v;
    s += v;
    q += v * v;
  }
  atomicAdd(&st[ccol], s);
  atomicAdd(&st[64 + ccol], q);
}

// ---------------- K2: GCN layer 1 (one (b,t) slice per block) ----------------
// z1 = adjn[b] @ x[b,t] @ W1 + b1 ; accumulate BN1 partial stats.
__global__ __launch_bounds__(128) void k_gcn1(
    const float* __restrict__ x, const float* __restrict__ adjn,
    const float* __restrict__ w1, const float* __restrict__ b1,
    float* __restrict__ z1, float* __restrict__ statp) {
  __shared__ __align__(16) float sAT[32 * STR64];  // adj^T  (A-layout, K=64,M=64)
  __shared__ __align__(16) float sX[32 * STR16];   // x      (B-layout, K=64,N=16)
  __shared__ __align__(16) float sYT[8 * STR64];   // Y^T    (A-layout, K=16,M=64)
  __shared__ __align__(16) float sW[8 * STR64];    // W1     (B-layout, K=16,N=64)
  const int tid = threadIdx.x, lane = tid & 31, wid = tid >> 5;
  const int bt = blockIdx.x;
  const float* Ab = adjn + (bt >> 9) * 4096;
  const float* Xs = x + (size_t)bt * 1024;

  // adj^T: element (k=w, m=v)
#pragma unroll
  for (int j = 0; j < 8; ++j) {
    const int i4 = (j << 9) + (tid << 2);
    const float4 a4 = *(const float4*)(Ab + i4);
    const int v = i4 >> 6, w0 = i4 & 63;
    sAT[((w0 + 0) >> 1) * STR64 + 2 * v + ((w0 + 0) & 1)] = a4.x;
    sAT[((w0 + 1) >> 1) * STR64 + 2 * v + ((w0 + 1) & 1)] = a4.y;
    sAT[((w0 + 2) >> 1) * STR64 + 2 * v + ((w0 + 2) & 1)] = a4.z;
    sAT[((w0 + 3) >> 1) * STR64 + 2 * v + ((w0 + 3) & 1)] = a4.w;
  }
  // X: element (k=w, n=c)
#pragma unroll
  for (int j = 0; j < 2; ++j) {
    const int i4 = (j << 9) + (tid << 2);
    const float4 x4 = *(const float4*)(Xs + i4);
    const int w = i4 >> 4, c0 = i4 & 15;
    const int base = (w >> 1) * STR16 + (w & 1);
    sX[base + 2 * (c0 + 0)] = x4.x;
    sX[base + 2 * (c0 + 1)] = x4.y;
    sX[base + 2 * (c0 + 2)] = x4.z;
    sX[base + 2 * (c0 + 3)] = x4.w;
  }
  // W1: element (k, n)
#pragma unroll
  for (int j = 0; j < 2; ++j) {
    const int i4 = (j << 9) + (tid << 2);
    const float4 w4 = *(const float4*)(w1 + i4);
    const int k = i4 >> 6, n0 = i4 & 63;
    const int base = (k >> 1) * STR64 + (k & 1);
    sW[base + 2 * (n0 + 0)] = w4.x;
    sW[base + 2 * (n0 + 1)] = w4.y;
    sW[base + 2 * (n0 + 2)] = w4.z;
    sW[base + 2 * (n0 + 3)] = w4.w;
  }
  __syncthreads();

  const int M0 = wid << 4;
  const int mrow = M0 + (lane & 15);
  const int col = lane & 15;
  const int half = lane >> 4;  // kh = 2*half
  const int rbase = M0 + (half << 3);

  // Y[64,16] = A @ X   (K=64, one N tile)
  const float* aB = sAT + half * STR64 + 2 * mrow;
  const float* xB = sX + half * STR16 + 2 * col;
  v8f acc = {};
#pragma unroll
  for (int k0 = 0; k0 < 64; k0 += 4) {
    const v2f a = *(const v2f*)(aB + (k0 >> 1) * STR64);
    const v2f b = *(const v2f*)(xB + (k0 >> 1) * STR16);
    acc = wmma4(a, b, acc);
  }
  // store Y^T: element (k=col, m=row)
  {
    const int yb = (col >> 1) * STR64 + (col & 1) + 2 * rbase;
#pragma unroll
    for (int r = 0; r < 8; ++r) sYT[yb + 2 * r] = acc[r];
  }
  __syncthreads();

  // Z[64,64] = Y @ W1 + b1   (K=16, 4 N tiles)
  v8f c0, c1, c2, c3;
  {
    const float v0 = b1[col], v1 = b1[16 + col], v2 = b1[32 + col], v3 = b1[48 + col];
#pragma unroll
    for (int r = 0; r < 8; ++r) { c0[r] = v0; c1[r] = v1; c2[r] = v2; c3[r] = v3; }
  }
  const float* yB = sYT + half * STR64 + 2 * mrow;
  const float* wB = sW + half * STR64 + 2 * col;
#pragma unroll
  for (int k0 = 0; k0 < 16; k0 += 4) {
    const int off = (k0 >> 1) * STR64;
    const v2f a = *(const v2f*)(yB + off);
    v2f b;
    b = *(const v2f*)(wB + off);      c0 = wmma4(a, b, c0);
    b = *(const v2f*)(wB + off + 32); c1 = wmma4(a, b, c1);
    b = *(const v2f*)(wB + off + 64); c2 = wmma4(a, b, c2);
    b = *(const v2f*)(wB + off + 96); c3 = wmma4(a, b, c3);
  }
  float* zs = z1 + (size_t)bt * 4096;
  float* st = statp + (bt & 63) * 128;
  emit_tile(c0, zs, st, rbase, col);
  emit_tile(c1, zs, st, rbase, 16 + col);
  emit_tile(c2, zs, st, rbase, 32 + col);
  emit_tile(c3, zs, st, rbase, 48 + col);
}

// ---------------- K3/K5: fold stats into (scale, shift) ----------------
__global__ __launch_bounds__(64) void k_finstat(
    const float* __restrict__ part, const float* __restrict__ g,
    const float* __restrict__ be, float* __restrict__ bn, float invn) {
  const int c = threadIdx.x;
  float s = 0.f, q = 0.f;
  for (int k = 0; k < 64; ++k) { s += part[k * 128 + c]; q += part[k * 128 + 64 + c]; }
  const float mean = s * invn;
  const float var = q * invn - mean * mean;
  const float a = g[c] * rsqrtf(var + 1e-5f);
  bn[c] = a;
  bn[64 + c] = be[c] - mean * a;
}

// ---------------- K4: GCN layer 2 ----------------
// x1 = relu(bn1(z1)); z2 = adjn[b] @ x1 @ W2 + b2 ; BN2 partial stats.
__global__ __launch_bounds__(128) void k_gcn2(
    const float* __restrict__ adjn, const float* __restrict__ z1,
    const float* __restrict__ w2, const float* __restrict__ b2,
    const float* __restrict__ bn1, float* __restrict__ z2,
    float* __restrict__ statp) {
  __shared__ __align__(16) float sAW[32 * STR64];  // adj^T (A-layout), then W2 (B-layout)
  __shared__ __align__(16) float sX[32 * STR64];   // x1    (B-layout)
  __shared__ __align__(16) float sYT[32 * STR64];  // Y^T   (A-layout)
  const int tid = threadIdx.x, lane = tid & 31, wid = tid >> 5;
  const int bt = blockIdx.x;
  const float* Ab = adjn + (bt >> 9) * 4096;
  const float* z1s = z1 + (size_t)bt * 4096;

  // x1 = relu(bn1(z1)) -> B-layout (k=w, n=c)
#pragma unroll
  for (int j = 0; j < 8; ++j) {
    const int i4 = (j << 9) + (tid << 2);
    const float4 zv = *(const float4*)(z1s + i4);
    const int w = i4 >> 6, c0 = i4 & 63;
    const int base = (w >> 1) * STR64 + (w & 1);
    sX[base + 2 * (c0 + 0)] = fmaxf(fmaf(zv.x, bn1[c0 + 0], bn1[64 + c0 + 0]), 0.f);
    sX[base + 2 * (c0 + 1)] = fmaxf(fmaf(zv.y, bn1[c0 + 1], bn1[64 + c0 + 1]), 0.f);
    sX[base + 2 * (c0 + 2)] = fmaxf(fmaf(zv.z, bn1[c0 + 2], bn1[64 + c0 + 2]), 0.f);
    sX[base + 2 * (c0 + 3)] = fmaxf(fmaf(zv.w, bn1[c0 + 3], bn1[64 + c0 + 3]), 0.f);
  }
  // adj^T: element (k=w, m=v)
#pragma unroll
  for (int j = 0; j < 8; ++j) {
    const int i4 = (j << 9) + (tid << 2);
    const float4 a4 = *(const float4*)(Ab + i4);
    const int v = i4 >> 6, w0 = i4 & 63;
    sAW[((w0 + 0) >> 1) * STR64 + 2 * v + ((w0 + 0) & 1)] = a4.x;
    sAW[((w0 + 1) >> 1) * STR64 + 2 * v + ((w0 + 1) & 1)] = a4.y;
    sAW[((w0 + 2) >> 1) * STR64 + 2 * v + ((w0 + 2) & 1)] = a4.z;
    sAW[((w0 + 3) >> 1) * STR64 + 2 * v + ((w0 + 3) & 1)] = a4.w;
  }
  __syncthreads();

  const int M0 = wid << 4;
  const int mrow = M0 + (lane & 15);
  const int col = lane & 15;
  const int half = lane >> 4;
  const int rbase = M0 + (half << 3);

  // Y[64,64] = A @ X1  (K=64, 4 N tiles)
  const float* aB = sAW + half * STR64 + 2 * mrow;
  const float* xB = sX + half * STR64 + 2 * col;
  v8f c0 = {}, c1 = {}, c2 = {}, c3 = {};
#pragma unroll
  for (int k0 = 0; k0 < 64; k0 += 4) {
    const int off = (k0 >> 1) * STR64;
    const v2f a = *(const v2f*)(aB + off);
    v2f b;
    b = *(const v2f*)(xB + off);      c0 = wmma4(a, b, c0);
    b = *(const v2f*)(xB + off + 32); c1 = wmma4(a, b, c1);
    b = *(const v2f*)(xB + off + 64); c2 = wmma4(a, b, c2);
    b = *(const v2f*)(xB + off + 96); c3 = wmma4(a, b, c3);
  }
  // store Y^T: element (k = n0+col, m = row); tiles are 8*STR64 apart
  {
    const int yb = (col >> 1) * STR64 + (col & 1) + 2 * rbase;
#pragma unroll
    for (int r = 0; r < 8; ++r) {
      sYT[yb + 2 * r] = c0[r];
      sYT[yb + 8 * STR64 + 2 * r] = c1[r];
      sYT[yb + 16 * STR64 + 2 * r] = c2[r];
      sYT[yb + 24 * STR64 + 2 * r] = c3[r];
    }
  }
  __syncthreads();
  // reuse sAW for W2 (B-layout)
#pragma unroll
  for (int j = 0; j < 8; ++j) {
    const int i4 = (j << 9) + (tid << 2);
    const float4 w4 = *(const float4*)(w2 + i4);
    const int k = i4 >> 6, n0 = i4 & 63;
    const int base = (k >> 1) * STR64 + (k & 1);
    sAW[base + 2 * (n0 + 0)] = w4.x;
    sAW[base + 2 * (n0 + 1)] = w4.y;
    sAW[base + 2 * (n0 + 2)] = w4.z;
    sAW[base + 2 * (n0 + 3)] = w4.w;
  }
  __syncthreads();

  // Z[64,64] = Y @ W2 + b2  (K=64, 4 N tiles)
  {
    const float v0 = b2[col], v1 = b2[16 + col], v2 = b2[32 + col], v3 = b2[48 + col];
#pragma unroll
    for (int r = 0; r < 8; ++r) { c0[r] = v0; c1[r] = v1; c2[r] = v2; c3[r] = v3; }
  }
  const float* yB = sYT + half * STR64 + 2 * mrow;
  const float* wB = sAW + half * STR64 + 2 * col;
#pragma unroll
  for (int k0 = 0; k0 < 64; k0 += 4) {
    const int off = (k0 >> 1) * STR64;
    const v2f a = *(const v2f*)(yB + off);
    v2f b;
    b = *(const v2f*)(wB + off);      c0 = wmma4(a, b, c0);
    b = *(const v2f*)(wB + off + 32); c1 = wmma4(a, b, c1);
    b = *(const v2f*)(wB + off + 64); c2 = wmma4(a, b, c2);
    b = *(const v2f*)(wB + off + 96); c3 = wmma4(a, b, c3);
  }
  float* zs = z2 + (size_t)bt * 4096;
  float* st = statp + (bt & 63) * 128;
  emit_tile(c0, zs, st, rbase, col);
  emit_tile(c1, zs, st, rbase, 16 + col);
  emit_tile(c2, zs, st, rbase, 32 + col);
  emit_tile(c3, zs, st, rbase, 48 + col);
}

// ---------------- K6: x12 = relu(bn1(z1)) + relu(bn2(z2)); reduce over T ----
__global__ __launch_bounds__(256) void k_reduceT(
    const float* __restrict__ z1, const float* __restrict__ z2,
    const float* __restrict__ bn1, const float* __restrict__ bn2,
    float* __restrict__ S, float* __restrict__ F, float* __restrict__ L) {
  const int b = blockIdx.x >> 5;
  const int chunk = blockIdx.x & 31;
  const int tid = threadIdx.x;
  const int cb = (tid << 2) & 63;  // channel base, same for all j
  float a1[4], d1[4], a2[4], d2[4];
#pragma unroll
  for (int u = 0; u < 4; ++u) {
    a1[u] = bn1[cb + u]; d1[u] = bn1[64 + cb + u];
    a2[u] = bn2[cb + u]; d2[u] = bn2[64 + cb + u];
  }
  float4 acc[4];
#pragma unroll
  for (int j = 0; j < 4; ++j) acc[j] = make_float4(0.f, 0.f, 0.f, 0.f);
  for (int tt = 0; tt < 16; ++tt) {
    const int t = (chunk << 4) + tt;
    const size_t base = ((size_t)((b << 9) + t)) << 12;
#pragma unroll
    for (int j = 0; j < 4; ++j) {
      const int vc = (j << 10) + (tid << 2);
      const float4 u1 = *(const float4*)(z1 + base + vc);
      const float4 u2 = *(const float4*)(z2 + base + vc);
      float4 sv;
      sv.x = fmaxf(fmaf(u1.x, a1[0], d1[0]), 0.f) + fmaxf(fmaf(u2.x, a2[0], d2[0]), 0.f);
      sv.y = fmaxf(fmaf(u1.y, a1[1], d1[1]), 0.f) + fmaxf(fmaf(u2.y, a2[1], d2[1]), 0.f);
      sv.z = fmaxf(fmaf(u1.z, a1[2], d1[2]), 0.f) + fmaxf(fmaf(u2.z, a2[2], d2[2]), 0.f);
      sv.w = fmaxf(fmaf(u1.w, a1[3], d1[3]), 0.f) + fmaxf(fmaf(u2.w, a2[3], d2[3]), 0.f);
      acc[j].x += sv.x; acc[j].y += sv.y; acc[j].z += sv.z; acc[j].w += sv.w;
      if (t == 0) *(float4*)(F + (b << 12) + vc) = sv;
      if (t == 511) *(float4*)(L + (b << 12) + vc) = sv;
    }
  }
#pragma unroll
  for (int j = 0; j < 4; ++j) {
    const int vc = (j << 10) + (tid << 2);
    atomicAdd(&S[(b << 12) + vc + 0], acc[j].x);
    atomicAdd(&S[(b << 12) + vc + 1], acc[j].y);
    atomicAdd(&S[(b << 12) + vc + 2], acc[j].z);
    atomicAdd(&S[(b << 12) + vc + 3], acc[j].w);
  }
}

// ---------------- K7: collapsed conv+mean, then MLP head ----------------
__global__ __launch_bounds__(256) void k_head(
    const float* __restrict__ S, const float* __restrict__ F,
    const float* __restrict__ L, const float* __restrict__ tw,
    const float* __restrict__ tb, const float* __restrict__ f1w,
    const float* __restrict__ f1b, const float* __restrict__ f2w,
    const float* __restrict__ f2b, float* __restrict__ out) {
  __shared__ __align__(16) float sS[4096], sF[4096], sL[4096];
  __shared__ float ysh[256], hsh[128];
  const int b = blockIdx.x, o = threadIdx.x;
#pragma unroll
  for (int j = 0; j < 4; ++j) {
    const int i4 = (j << 10) + (o << 2);
    *(float4*)(sS + i4) = *(const float4*)(S + b * 4096 + i4);
    *(float4*)(sF + i4) = *(const float4*)(F + b * 4096 + i4);
    *(float4*)(sL + i4) = *(const float4*)(L + b * 4096 + i4);
  }
  __syncthreads();
  float acc = 0.f;
  const float* twr = tw + (size_t)o * 4096 * 3;
  for (int i = 0; i < 4096; ++i) {
    const float s = sS[i];
    acc += twr[i * 3 + 0] * (s - sL[i]) + twr[i * 3 + 1] * s +
           twr[i * 3 + 2] * (s - sF[i]);
  }
  ysh[o] = acc * (1.0f / 512.0f) + tb[o];
  __syncthreads();
  if (o < 128) {
    float h = f1b[o];
    for (int q = 0; q < 256; ++q) h = fmaf(ysh[q], f1w[q * 128 + o], h);
    hsh[o] = fmaxf(h, 0.f);
  }
  __syncthreads();
  if (o < 10) {
    float r = f2b[o];
    for (int p = 0; p < 128; ++p) r = fmaf(hsh[p], f2w[p * 10 + o], r);
    out[b * 10 + o] = r;
  }
}

extern "C" void kernel_launch(void* const* d_in, const int* in_sizes, int n_in,
                              void* d_out, int out_size, void* d_ws, size_t ws_size,
                              hipStream_t stream) {
  (void)in_sizes; (void)n_in; (void)out_size; (void)ws_size;
  const float* x   = (const float*)d_in[0];
  const float* adj = (const float*)d_in[1];
  const float* w1  = (const float*)d_in[2];
  const float* b1  = (const float*)d_in[3];
  const float* g1  = (const float*)d_in[4];
  const float* be1 = (const float*)d_in[5];
  const float* w2  = (const float*)d_in[6];
  const float* b2  = (const float*)d_in[7];
  const float* g2  = (const float*)d_in[8];
  const float* be2 = (const float*)d_in[9];
  const float* tw  = (const float*)d_in[10];
  const float* tb  = (const float*)d_in[11];
  const float* f1w = (const float*)d_in[12];
  const float* f1b = (const float*)d_in[13];
  const float* f2w = (const float*)d_in[14];
  const float* f2b = (const float*)d_in[15];
  float* out = (float*)d_out;
  float* ws  = (float*)d_ws;

  float* adjn  = ws + OFF_ADJN;
  float* stat1 = ws + OFF_STAT1;
  float* stat2 = ws + OFF_STAT2;
  float* bn1   = ws + OFF_BN1;
  float* bn2   = ws + OFF_BN2;
  float* Sbuf  = ws + OFF_S;
  float* Fbuf  = ws + OFF_F;
  float* Lbuf  = ws + OFF_L;
  float* z1    = ws + OFF_Z1;
  float* z2    = ws + OFF_Z2;

  const float invn = 1.0f / 524288.0f;  // B*T*V

  k_zero<<<321, 256, 0, stream>>>(stat1, 82176);  // stats + bn + S regions
  k_adj<<<16, 64, 0, stream>>>(adj, adjn);
  k_gcn1<<<8192, 128, 0, stream>>>(x, adjn, w1, b1, z1, stat1);
  k_finstat<<<1, 64, 0, stream>>>(stat1, g1, be1, bn1, invn);
  k_gcn2<<<8192, 128, 0, stream>>>(adjn, z1, w2, b2, bn1, z2, stat2);
  k_finstat<<<1, 64, 0, stream>>>(stat2, g2, be2, bn2, invn);
  k_reduceT<<<512, 256, 0, stream>>>(z1, z2, bn1, bn2, Sbuf, Fbuf, Lbuf);
  k_head<<<16, 256, 0, stream>>>(Sbuf, Fbuf, Lbuf, tw, tb, f1w, f1b, f2w, f2b, out);
}